// Sae_81449759801981
// MI455X (gfx1250) — compile-verified
//
#include <hip/hip_runtime.h>

// Problem constants (fixed by the reference harness)
#define DD   768
#define LL   12288
#define TOPK 20
#define QCAP 256

typedef __attribute__((ext_vector_type(2))) float v2f;
typedef __attribute__((ext_vector_type(4))) float v4f;
typedef __attribute__((ext_vector_type(8))) float v8f;

#define XPITCH 772   // 768 + 4 pad words -> conflict-free A-operand ds_load_b64
#define TPITCH 68    // tile pitch: 16B-aligned v4f rows for the filter phase

// ---------------------------------------------------------------------------
// Kernel 0: transpose W_dec [768, 12288] -> W_decT [12288, 768] into scratch
// ---------------------------------------------------------------------------
__global__ __launch_bounds__(256) void sae_transpose_wdec(
    const float* __restrict__ in, float* __restrict__ outT)
{
  __shared__ float t[32][33];
  const int bx = blockIdx.x;           // along L (12288/32 = 384)
  const int by = blockIdx.y;           // along D (768/32  = 24)
  const int tx = threadIdx.x;          // 0..31
  const int ty = threadIdx.y;          // 0..7
#pragma unroll
  for (int i = 0; i < 32; i += 8)
    t[ty + i][tx] = in[(size_t)(by * 32 + ty + i) * LL + bx * 32 + tx];
  __syncthreads();
#pragma unroll
  for (int i = 0; i < 32; i += 8)
    outT[(size_t)(bx * 32 + ty + i) * DD + by * 32 + tx] = t[tx][ty + i];
}

// ---------------------------------------------------------------------------
// Kernel 1: fused encoder (f32 WMMA, 4 accumulator chains) + parallel
// threshold-filtered streaming top-20 + sparse decoder.
// One workgroup = 128 threads = 4 waves, owns 16 rows of x.
// Each wave computes a 16x16 latents tile per L-step (64 latents / step).
// ---------------------------------------------------------------------------
__global__ __launch_bounds__(128) void sae_fused(
    const float* __restrict__ x,     const float* __restrict__ Wenc,
    const float* __restrict__ benc,  const float* __restrict__ Wdec,
    const float* __restrict__ bdec,  const float* __restrict__ WdecT,
    float* __restrict__ out, int useT)
{
  __shared__ float lds_x[16 * XPITCH];        // 49408 B  staged x block
  __shared__ float tile[16 * TPITCH];         //  4352 B  latents tile
  __shared__ float top_val[16][TOPK + 1];     //  per-row top-k values
  __shared__ int   top_idx[16][TOPK + 1];     //  per-row top-k indices
  __shared__ float lds_min[16];               //  per-row threshold
  __shared__ int   qcount;                    //  candidate queue counter
  __shared__ float qval[QCAP];
  __shared__ int   qkey[QCAP];                //  (row << 16) | latent_index

  const int tid    = threadIdx.x;
  const int wave   = tid >> 5;
  const int lane   = tid & 31;
  const int lane16 = lane & 15;
  const int hi     = lane >> 4;      // 0: K lanes {0,1}; 1: K lanes {2,3}
  const int khalf  = hi << 1;
  const int row_base = blockIdx.x * 16;
  const float NEG_INF = -__builtin_inff();

  // ---- stage 16 rows of x into LDS (coalesced, one-time) ----
  for (int i = tid; i < 16 * DD; i += 128) {
    const int r = i / DD;
    const int c = i - r * DD;
    lds_x[r * XPITCH + c] = x[(size_t)(row_base + r) * DD + c];
  }
  if (tid < 16) {
#pragma unroll
    for (int s = 0; s < TOPK; ++s) { top_val[tid][s] = NEG_INF; top_idx[tid][s] = 0; }
    lds_min[tid] = NEG_INF;
  }
  __syncthreads();

  float cur_min = NEG_INF;   // live in owner-thread (tid<16) registers
  int   min_pos = 0;

  auto insert = [&](float val, int idx) {
    top_val[tid][min_pos] = val;
    top_idx[tid][min_pos] = idx;
    float mm = top_val[tid][0]; int mp = 0;
#pragma unroll
    for (int s = 1; s < TOPK; ++s) {
      const float t = top_val[tid][s];
      if (t < mm) { mm = t; mp = s; }
    }
    cur_min = mm; min_pos = mp;
  };

  // A operand: lane l<16 holds x[row=l, k..k+1]; lane 16+l holds x[row=l, k+2..k+3]
  const float* __restrict__ xl = lds_x + lane16 * XPITCH + khalf;

  // filter-phase assignment: 8 threads per row, 8 candidates per thread
  const int frow = tid >> 3;
  const int fcol = (tid & 7) * 8;
  const float* __restrict__ trow = tile + frow * TPITCH + fcol;

  for (int lt = 0; lt < LL; lt += 64) {
    const int col0 = lt + wave * 16;                    // this wave's 16 latents
    // B operand: B[k,n] = Wenc[col0+n, d0+k] -> contiguous 8B per lane
    const float* __restrict__ wr = Wenc + (size_t)(col0 + lane16) * DD + khalf;

    // ---- encoder tile: 192 f32 WMMAs on 4 independent accumulator chains ----
    v8f c0 = {}; v8f c1 = {}; v8f c2 = {}; v8f c3 = {};
#pragma unroll 2
    for (int k = 0; k < DD; k += 16) {
      v2f a0 = *(const v2f*)(xl + k);      v2f b0 = *(const v2f*)(wr + k);
      v2f a1 = *(const v2f*)(xl + k + 4);  v2f b1 = *(const v2f*)(wr + k + 4);
      v2f a2 = *(const v2f*)(xl + k + 8);  v2f b2 = *(const v2f*)(wr + k + 8);
      v2f a3 = *(const v2f*)(xl + k + 12); v2f b3 = *(const v2f*)(wr + k + 12);
      c0 = __builtin_amdgcn_wmma_f32_16x16x4_f32(false, a0, false, b0, (short)0, c0, false, false);
      c1 = __builtin_amdgcn_wmma_f32_16x16x4_f32(false, a1, false, b1, (short)0, c1, false, false);
      c2 = __builtin_amdgcn_wmma_f32_16x16x4_f32(false, a2, false, b2, (short)0, c2, false, false);
      c3 = __builtin_amdgcn_wmma_f32_16x16x4_f32(false, a3, false, b3, (short)0, c3, false, false);
    }

    __syncthreads();   // (A) previous step's filter/drain done with tile & queue

    const float bias = benc[col0 + lane16];
    const int moff = hi << 3;   // C/D layout: vgpr v -> row v (lanes 0-15) / v+8 (16-31)
#pragma unroll
    for (int v = 0; v < 8; ++v)
      tile[(moff + v) * TPITCH + wave * 16 + lane16] =
          (c0[v] + c1[v]) + (c2[v] + c3[v]) + bias;
    if (tid == 0) qcount = 0;

    __syncthreads();   // (B) tile + reset counter visible

    // ---- parallel threshold filter: 1024 candidates / 128 threads ----
    {
      const float thr = lds_min[frow];
      v4f va = *(const v4f*)(trow);
      v4f vb = *(const v4f*)(trow + 4);
      const float ma = fmaxf(fmaxf(va[0], va[1]), fmaxf(va[2], va[3]));
      const float mb = fmaxf(fmaxf(vb[0], vb[1]), fmaxf(vb[2], vb[3]));
      if (fmaxf(ma, mb) > thr) {                    // rare after warm-up
#pragma unroll
        for (int i = 0; i < 8; ++i) {
          const float v = (i < 4) ? va[i] : vb[i - 4];
          if (v > thr) {
            const int slot = atomicAdd(&qcount, 1);
            if (slot < QCAP) {
              qval[slot] = v;
              qkey[slot] = (frow << 16) | (lt + fcol + i);
            }
          }
        }
      }
    }

    __syncthreads();   // (C) queue complete; wave0 drains while others compute on

    if (tid < 16) {
      const int nq = qcount;
      if (nq > QCAP) {
        // overflow (first step(s) only): exclusive full scan of my row
        for (int cx = 0; cx < 64; ++cx) {
          const float val = tile[tid * TPITCH + cx];
          if (val > cur_min) insert(val, lt + cx);
        }
      } else {
        for (int i = 0; i < nq; ++i) {
          const int key = qkey[i];              // broadcast LDS read
          if ((key >> 16) == tid) {
            const float val = qval[i];
            if (val > cur_min) insert(val, key & 0xffff);
          }
        }
      }
      lds_min[tid] = cur_min;
    }
  }

  __syncthreads();     // final drain visible before decode

  // ---- sparse decoder: out[row, :] = b_dec + sum_j val_j * Wdec[:, idx_j] ----
  for (int r = 0; r < 16; ++r) {
    const size_t orow = (size_t)(row_base + r) * DD;
    if (useT) {
      for (int d = tid; d < DD; d += 128) {
        float acc = bdec[d];
#pragma unroll
        for (int s = 0; s < TOPK; ++s)
          acc += top_val[r][s] * WdecT[(size_t)top_idx[r][s] * DD + d];
        out[orow + d] = acc;
      }
    } else {
      for (int d = tid; d < DD; d += 128) {
        float acc = bdec[d];
#pragma unroll
        for (int s = 0; s < TOPK; ++s)
          acc += top_val[r][s] * Wdec[(size_t)d * LL + top_idx[r][s]];
        out[orow + d] = acc;
      }
    }
  }
}

// ---------------------------------------------------------------------------
extern "C" void kernel_launch(void* const* d_in, const int* in_sizes, int n_in,
                              void* d_out, int out_size, void* d_ws, size_t ws_size,
                              hipStream_t stream)
{
  const float* x    = (const float*)d_in[0];
  const float* Wenc = (const float*)d_in[1];
  const float* benc = (const float*)d_in[2];
  const float* Wdec = (const float*)d_in[3];
  const float* bdec = (const float*)d_in[4];
  float* out = (float*)d_out;

  const int N = in_sizes[0] / DD;              // 16384
  const size_t needT = (size_t)LL * DD * sizeof(float);
  const int useT = (ws_size >= needT) ? 1 : 0;
  float* WdecT = (float*)d_ws;

  if (useT) {
    dim3 g(LL / 32, DD / 32), b(32, 8);
    sae_transpose_wdec<<<g, b, 0, stream>>>(Wdec, WdecT);
  }
  sae_fused<<<N / 16, 128, 0, stream>>>(x, Wenc, benc, Wdec, bdec, WdecT, out, useT);
}